// MSE_full_26706106646779
// MI455X (gfx1250) — compile-verified
//
#include <hip/hip_runtime.h>
#include <math.h>

// CDNA5 / gfx1250: wave32, WMMA 16x16 shapes, async global->LDS staging.
typedef __attribute__((ext_vector_type(2))) float v2f;
typedef __attribute__((ext_vector_type(8))) float v8f;

#define RAD   4
#define WINSZ 9
#define TD    16            // output tile edge
#define PD    24            // TD + 2*RAD (haloed tile edge)
#define SEGS  6             // 24 floats per row / 4 floats per b128 segment

#define DIM_D 160
#define DIM_H 192
#define DIM_W 160
#define NB    2
#define N_TOTAL (2.0f * 160.0f * 192.0f * 160.0f)   // 9,830,400

__device__ __forceinline__ int clampi(int v, int lo, int hi) {
    return v < lo ? lo : (v > hi ? hi : v);
}

// Fused kernel: async-stage haloed I/J tiles to LDS, diff = I^2 - J^2,
// separable 9^3 box-sum in LDS (H-pass on the WMMA pipe), partial sum of
// (1 - |s|/729) per block.
__global__ __launch_bounds__(256)
void ncc_box_kernel(const float* __restrict__ y_pred,
                    const float* __restrict__ y_true,
                    float* __restrict__ partial) {
    __shared__ float bufA[PD * PD * PD];   // raw I -> diff -> s2   (54 KB)
    __shared__ float bufB[PD * PD * PD];   // raw J                 (54 KB)
    __shared__ float s1[PD * PD * TD];     // after W-pass          (36 KB)
    __shared__ float red[256];
    float* s2 = bufA;                      // [z][yo][x] after H-pass (alias)

    const int tid = threadIdx.x;
    const int b   = blockIdx.z / (DIM_D / TD);
    const int tz  = blockIdx.z % (DIM_D / TD);
    const int z0  = tz * TD - RAD;
    const int y0  = blockIdx.y * TD - RAD;
    const int x0  = blockIdx.x * TD - RAD;

    // LDS byte offsets: generic-pointer low 32 bits == LDS byte address.
    const unsigned ldsA = (unsigned)(unsigned long long)&bufA[0];
    const unsigned ldsB = (unsigned)(unsigned long long)&bufB[0];

    // ---- Step 1a: async-stage both haloed tiles into LDS (b128/lane) ----
    // OOB segments use clamped (in-tensor) addresses; zero-masked in 1b.
    for (int seg = tid; seg < PD * PD * SEGS; seg += 256) {
        const int sx = seg % SEGS;
        const int ly = (seg / SEGS) % PD;
        const int lz = seg / (SEGS * PD);
        const int gz = clampi(z0 + lz, 0, DIM_D - 1);
        const int gy = clampi(y0 + ly, 0, DIM_H - 1);
        const int gx = clampi(x0 + 4 * sx, 0, DIM_W - 4);
        const long idx = (((long)b * DIM_D + gz) * DIM_H + gy) * DIM_W + gx;
        const unsigned loff = (unsigned)(((lz * PD + ly) * PD + 4 * sx) * 4);
        const unsigned long long ga = (unsigned long long)(y_true + idx);
        const unsigned long long gb = (unsigned long long)(y_pred + idx);
        asm volatile("global_load_async_to_lds_b128 %0, %1, off"
                     :: "v"(ldsA + loff), "v"(ga) : "memory");
        asm volatile("global_load_async_to_lds_b128 %0, %1, off"
                     :: "v"(ldsB + loff), "v"(gb) : "memory");
    }
    asm volatile("s_wait_asynccnt 0" ::: "memory");
    __syncthreads();

    // ---- Step 1b: diff = I^2 - J^2 in place, zero padding mask ----
    for (int l = tid; l < PD * PD * PD; l += 256) {
        const int lx = l % PD;
        const int ly = (l / PD) % PD;
        const int lz = l / (PD * PD);
        const int gz = z0 + lz, gy = y0 + ly, gx = x0 + lx;
        const bool ok = (gz >= 0 && gz < DIM_D && gy >= 0 && gy < DIM_H &&
                         gx >= 0 && gx < DIM_W);
        const float i = bufA[l];
        const float j = bufB[l];
        bufA[l] = ok ? (i * i - j * j) : 0.0f;
    }
    __syncthreads();

    // ---- Step 2: box-sum along W, sliding window over 24*24 rows ----
    for (int r = tid; r < PD * PD; r += 256) {
        const float* src = &bufA[r * PD];
        float* dst = &s1[r * TD];
        float s = 0.0f;
        #pragma unroll
        for (int k = 0; k < WINSZ; ++k) s += src[k];
        dst[0] = s;
        #pragma unroll
        for (int k = 1; k < TD; ++k) {
            s += src[k + WINSZ - 1] - src[k - 1];
            dst[k] = s;
        }
    }
    __syncthreads();

    // ---- Step 3: box-sum along H as banded GEMM on the matrix pipe ----
    // Out(16x16)[yo][x] = sum_k Band[yo][k] * s1[z][k][x], K = 24 = 6 x 4
    // via V_WMMA_F32_16X16X4_F32, one wave per z-slice (8 waves x 3 slices).
    {
        const int wave  = tid >> 5;
        const int lane  = tid & 31;
        const int m     = lane & 15;                // A: row yo;  B: col x
        const int khalf = (lane < 16) ? 0 : 2;      // 16x4 A / 4x16 B lane split
        for (int z = wave; z < PD; z += 8) {        // uniform per wave: EXEC all 1s
            v8f acc = {};
            #pragma unroll
            for (int c = 0; c < 6; ++c) {
                const int k0 = 4 * c + khalf;
                v2f a, bm;
                // Band[m][k] = 1 iff m <= k <= m+8  (9-wide window, zero-padded)
                a.x = (k0     >= m && k0     <= m + 8) ? 1.0f : 0.0f;
                a.y = (k0 + 1 >= m && k0 + 1 <= m + 8) ? 1.0f : 0.0f;
                bm.x = s1[(z * PD + k0    ) * TD + m];
                bm.y = s1[(z * PD + k0 + 1) * TD + m];
                acc = __builtin_amdgcn_wmma_f32_16x16x4_f32(
                    /*neg_a=*/false, a, /*neg_b=*/false, bm,
                    /*c_mod=*/(short)0, acc, /*reuse_a=*/false, /*reuse_b=*/false);
            }
            // C/D layout: VGPR v -> M = v + (lane<16 ? 0 : 8), N = lane%16
            const int mbase = (lane < 16) ? 0 : 8;
            #pragma unroll
            for (int v = 0; v < 8; ++v) {
                s2[(z * TD + (mbase + v)) * TD + m] = acc[v];
            }
        }
    }
    __syncthreads();

    // ---- Step 4: box-sum along D + fused 1 - |s|/729, partial reduce ----
    float local = 0.0f;
    for (int o = tid; o < TD * TD * TD; o += 256) {
        const int x  = o & 15;
        const int yo = (o >> 4) & 15;
        const int zo = o >> 8;
        float s = 0.0f;
        #pragma unroll
        for (int k = 0; k < WINSZ; ++k) s += s2[((zo + k) * TD + yo) * TD + x];
        local += 1.0f - fabsf(s) * (1.0f / 729.0f);
    }
    red[tid] = local;
    __syncthreads();
    for (int off = 128; off > 0; off >>= 1) {
        if (tid < off) red[tid] += red[tid + off];
        __syncthreads();
    }
    if (tid == 0) {
        const int blin = (blockIdx.z * gridDim.y + blockIdx.y) * gridDim.x + blockIdx.x;
        partial[blin] = red[0];
    }
}

// Deterministic final reduction: fixed traversal order, single block.
__global__ __launch_bounds__(256)
void ncc_reduce_kernel(const float* __restrict__ partial, float* __restrict__ out, int n) {
    __shared__ float red[256];
    float s = 0.0f;
    for (int i = threadIdx.x; i < n; i += 256) s += partial[i];
    red[threadIdx.x] = s;
    __syncthreads();
    for (int off = 128; off > 0; off >>= 1) {
        if (threadIdx.x < off) red[threadIdx.x] += red[threadIdx.x + off];
        __syncthreads();
    }
    if (threadIdx.x == 0) out[0] = red[0] * (1.0f / N_TOTAL);
}

extern "C" void kernel_launch(void* const* d_in, const int* in_sizes, int n_in,
                              void* d_out, int out_size, void* d_ws, size_t ws_size,
                              hipStream_t stream) {
    const float* y_pred = (const float*)d_in[0];   // setup_inputs order: y_pred, y_true
    const float* y_true = (const float*)d_in[1];
    float* out = (float*)d_out;
    float* partial = (float*)d_ws;                 // 2400 floats of scratch

    dim3 grid(DIM_W / TD, DIM_H / TD, NB * (DIM_D / TD));   // 10 x 12 x 20 = 2400
    ncc_box_kernel<<<grid, 256, 0, stream>>>(y_pred, y_true, partial);

    const int nblocks = grid.x * grid.y * grid.z;
    ncc_reduce_kernel<<<1, 256, 0, stream>>>(partial, out, nblocks);
}